// DropoutFBM_3_53300544143592
// MI455X (gfx1250) — compile-verified
//
#include <hip/hip_runtime.h>
#include <hip/hip_bf16.h>
#include <stdint.h>

// ---------------------------------------------------------------------------
// DropoutFBM on MI455X (gfx1250): memory-bound elementwise broadcast multiply.
// HBM floor: (128 MiB in + 128 MiB out) / 23.3 TB/s ~= 11.5 us. Mask setup is
// negligible (64K points, 64K cells). Streaming kernel uses b128 non-temporal
// global traffic and stages the reused mask tile into LDS with the gfx1250
// async-to-LDS path (ASYNCcnt + s_wait_asynccnt).
// ---------------------------------------------------------------------------

#define N_COL   256
#define N_ROW   256
#define N_CELLS (N_COL * N_ROW)   // 65536
#define BATCH   512

typedef float v4f __attribute__((ext_vector_type(4)));

// ---- ws layout -------------------------------------------------------------
// [0,           256KB)  float maskScaled[65536]
// [256KB,       264KB)  uint32 touchedBits[2048]  (65536-bit occupancy bitmap)
// [264KB,      264KB+4) int    touchedCount
#define WS_MASK_OFF  0
#define WS_BITS_OFF  (N_CELLS * 4)
#define WS_CNT_OFF   (WS_BITS_OFF + (N_CELLS / 8))

// ---------------------------------------------------------------------------
__global__ void zero_ws_kernel(uint32_t* __restrict__ bits, int* __restrict__ cnt) {
    int t = blockIdx.x * blockDim.x + threadIdx.x;
    if (t < N_CELLS / 32) bits[t] = 0u;
    if (t == 0) *cnt = 0;
}

// Scatter: each agent sample marks its cell if its fractional position lies in
// the [0.25, 0.75]^2 box. Matches reference fp32 ops (x*256, floorf, frac).
__global__ void touch_kernel(const float* __restrict__ ax,
                             const float* __restrict__ ay,
                             uint32_t* __restrict__ bits, int n) {
    int t = blockIdx.x * blockDim.x + threadIdx.x;
    if (t >= n) return;
    float fx = ax[t] * (float)N_COL;
    float fy = ay[t] * (float)N_ROW;
    float cx = floorf(fx);
    float cy = floorf(fy);
    float rx = fx - cx;
    float ry = fy - cy;
    bool in_box = (rx >= 0.25f) & (rx <= 0.75f) & (ry >= 0.25f) & (ry <= 0.75f);
    if (in_box) {
        int ix = min(max((int)cx, 0), N_COL - 1);
        int iy = min(max((int)cy, 0), N_ROW - 1);
        int cell = (ix << 8) | iy;               // ix * N_ROW + iy
        atomicOr(&bits[cell >> 5], 1u << (cell & 31));
    }
}

// One-block popcount reduction over the 2048-word bitmap (wave32-friendly tree).
__global__ void count_kernel(const uint32_t* __restrict__ bits, int* __restrict__ cnt) {
    __shared__ int red[256];
    int t = threadIdx.x;
    int s = 0;
    for (int i = t; i < N_CELLS / 32; i += 256) s += __popc(bits[i]);
    red[t] = s;
    __syncthreads();
    for (int w = 128; w > 0; w >>= 1) {
        if (t < w) red[t] += red[t + w];
        __syncthreads();
    }
    if (t == 0) *cnt = red[0];
}

// Build rot90(mask) premultiplied by scale. rot90 CCW: out[a,b] = m[b, 255-a],
// flat: maskScaled[a*256+b] = (touched[b*256 + 255 - a] ? 0 : scale).
// scale = 65536/sum(mask) = 65536/(65536 - touchedCount); correctly rounded,
// identical to the reference's 1/(1 - dropout_rate).
__global__ void build_mask_kernel(const uint32_t* __restrict__ bits,
                                  const int* __restrict__ cnt,
                                  float* __restrict__ maskScaled) {
    int k = blockIdx.x * blockDim.x + threadIdx.x;   // 0..65535
    int a = k >> 8;
    int b = k & 255;
    int cell = (b << 8) | (255 - a);
    uint32_t touched = (bits[cell >> 5] >> (cell & 31)) & 1u;
    float scale = 65536.0f / (float)(N_CELLS - *cnt);
    maskScaled[k] = touched ? 0.0f : scale;
}

// ---------------------------------------------------------------------------
// Streaming apply: grid (32 k-tiles, 64 row-groups), 256 threads.
// Each block: stage 2048 mask floats (8 KB) into LDS via async-to-LDS b128,
// then move 8 rows x 2048 cols with non-temporal b128 loads/stores so the
// 128 MB streams do not thrash L2 (mask array + LDS tiles stay hot).
// ---------------------------------------------------------------------------
#define TILE_K 2048
#define ROWS_PER_BLOCK 8

__global__ __launch_bounds__(256) void apply_kernel(const float* __restrict__ in,
                                                    const float* __restrict__ maskScaled,
                                                    float* __restrict__ out) {
    __shared__ float sm[TILE_K];

    const int t = threadIdx.x;
    const int kBase = blockIdx.x * TILE_K;
    const int rBase = blockIdx.y * ROWS_PER_BLOCK;

    // Async stage of the mask tile: 2 x b128 per thread -> 8 KB total.
    {
        uint32_t lds0 = (uint32_t)(uintptr_t)(&sm[t * 4]);
        uint64_t g0   = (uint64_t)(uintptr_t)(maskScaled + kBase + t * 4);
        asm volatile("global_load_async_to_lds_b128 %0, %1, off"
                     :: "v"(lds0), "v"(g0) : "memory");
        uint32_t lds1 = (uint32_t)(uintptr_t)(&sm[1024 + t * 4]);
        uint64_t g1   = (uint64_t)(uintptr_t)(maskScaled + kBase + 1024 + t * 4);
        asm volatile("global_load_async_to_lds_b128 %0, %1, off"
                     :: "v"(lds1), "v"(g1) : "memory");
        asm volatile("s_wait_asynccnt 0" ::: "memory");
    }
    __syncthreads();

    const v4f* smv = (const v4f*)sm;

#pragma unroll
    for (int r = 0; r < ROWS_PER_BLOCK; ++r) {
        const int row = rBase + r;
        const v4f* __restrict__ ip = (const v4f*)(in  + (size_t)row * N_CELLS + kBase);
        v4f* __restrict__       op = (v4f*)(out + (size_t)row * N_CELLS + kBase);
#pragma unroll
        for (int j = 0; j < TILE_K / (256 * 4); ++j) {   // 2 iterations
            const int idx = t + j * 256;                 // v4f index within tile
            v4f v = __builtin_nontemporal_load(ip + idx); // th:NT streaming load
            v4f m = smv[idx];                             // LDS b128 read
            v *= m;
            __builtin_nontemporal_store(v, op + idx);     // th:NT streaming store
        }
    }
}

// ---------------------------------------------------------------------------
extern "C" void kernel_launch(void* const* d_in, const int* in_sizes, int n_in,
                              void* d_out, int out_size, void* d_ws, size_t ws_size,
                              hipStream_t stream) {
    const float* inp = (const float*)d_in[0];   // [512, 65536] fp32
    const float* ax  = (const float*)d_in[1];   // [64, 1001]   fp32
    const float* ay  = (const float*)d_in[2];   // [64, 1001]   fp32
    float* out = (float*)d_out;                 // [512, 65536] fp32

    char* ws = (char*)d_ws;
    float*    maskScaled = (float*)(ws + WS_MASK_OFF);
    uint32_t* bits       = (uint32_t*)(ws + WS_BITS_OFF);
    int*      cnt        = (int*)(ws + WS_CNT_OFF);

    const int npts = in_sizes[1];               // 64 * 1001 = 64064

    // Scratch must be (re)initialized every call: harness does not re-poison.
    zero_ws_kernel<<<(N_CELLS / 32 + 255) / 256, 256, 0, stream>>>(bits, cnt);
    touch_kernel<<<(npts + 255) / 256, 256, 0, stream>>>(ax, ay, bits, npts);
    count_kernel<<<1, 256, 0, stream>>>(bits, cnt);
    build_mask_kernel<<<N_CELLS / 256, 256, 0, stream>>>(bits, cnt, maskScaled);

    dim3 grid(N_CELLS / TILE_K, BATCH / ROWS_PER_BLOCK);   // (32, 64)
    apply_kernel<<<grid, 256, 0, stream>>>(inp, maskScaled, out);
}